// ModernBertAttention_25331717112504
// MI455X (gfx1250) — compile-verified
//
#include <hip/hip_runtime.h>

typedef __attribute__((ext_vector_type(2))) float v2f;
typedef __attribute__((ext_vector_type(8))) float v8f;

#define BB   2
#define SS   2048
#define DD   768
#define HH   12
#define HDIM 64
#define WIN  64            // WINDOW/2
#define NEGV (-1000000000.0f)
#define KT   9             // key tiles per 16-query tile
#define KLEN (KT * 16)     // 144 candidate keys

// ---------------------------------------------------------------------------
// f32 WMMA: D = A(16x4) x B(4x16) + C   (V_WMMA_F32_16X16X4_F32)
// A frag (row-major MxK): lane l<16 -> A[r0+l][k+0..1];  l>=16 -> A[r0+l-16][k+2..3]
// B frag (row-major KxN): lane l<16 -> B[k+0..1][c0+l];  l>=16 -> B[k+2..3][c0+l-16]
// C/D: VGPR v, lane l<16 -> C[M=v][N=l]; lane l>=16 -> C[M=v+8][N=l-16]
// ---------------------------------------------------------------------------
__device__ inline v8f wmma_f32(v2f a, v2f b, v8f c) {
    return __builtin_amdgcn_wmma_f32_16x16x4_f32(
        false, a, false, b, (short)0, c, false, false);
}

// ---------------------------------------------------------------------------
// fp32 WMMA GEMM: C[M,N] = A[M,K] @ B[K,N], row-major.
// Wave computes 32x64 (MT=2 x NT=4); 4 waves stack on M -> 128x64 block tile.
// Requires M%128==0, N%64==0, K%4==0 (true for both GEMMs here).
// ---------------------------------------------------------------------------
template <int MT, int NT>
__global__ __launch_bounds__(128)
void wmma_gemm_f32(const float* __restrict__ A, const float* __restrict__ B,
                   float* __restrict__ C, int M, int N, int K) {
    const int lane = threadIdx.x & 31;
    const int wave = threadIdx.x >> 5;
    const int lr   = lane & 15;
    const int kh   = (lane >> 4) << 1;          // 0 or 2
    const int hi8  = (lane >> 4) << 3;          // 0 or 8

    const int n0 = blockIdx.x * (16 * NT);
    const int m0 = (blockIdx.y * 4 + wave) * (16 * MT);

    v8f acc[MT][NT];
#pragma unroll
    for (int i = 0; i < MT; ++i)
#pragma unroll
        for (int j = 0; j < NT; ++j) acc[i][j] = v8f{0,0,0,0,0,0,0,0};

    const float* ap[MT];
#pragma unroll
    for (int i = 0; i < MT; ++i)
        ap[i] = A + (size_t)(m0 + 16 * i + lr) * K + kh;
    const float* bp = B + (size_t)kh * N + n0 + lr;

    for (int k = 0; k < K; k += 4) {
        v2f a[MT];
#pragma unroll
        for (int i = 0; i < MT; ++i)
            a[i] = *(const v2f*)(ap[i] + k);
        v2f b[NT];
#pragma unroll
        for (int j = 0; j < NT; ++j) {
            const float* q = bp + (size_t)k * N + 16 * j;
            b[j] = v2f{q[0], q[N]};
        }
#pragma unroll
        for (int i = 0; i < MT; ++i)
#pragma unroll
            for (int j = 0; j < NT; ++j)
                acc[i][j] = wmma_f32(a[i], b[j], acc[i][j]);
    }

#pragma unroll
    for (int i = 0; i < MT; ++i)
#pragma unroll
        for (int j = 0; j < NT; ++j)
#pragma unroll
            for (int v = 0; v < 8; ++v)
                C[(size_t)(m0 + 16 * i + v + hi8) * N + n0 + 16 * j + lr] = acc[i][j][v];
}

// ---------------------------------------------------------------------------
// One 16x16 score tile: S^T[key][query] accumulated over HDIM.
// GUARD=false: all key rows provably in [0,S) -> raw b64 loads, no flag mul.
// ---------------------------------------------------------------------------
template <bool GUARD>
__device__ inline v8f score_tile(const float* __restrict__ kbase,
                                 const float* __restrict__ qp,
                                 int k0, int lr, int khalf, int row) {
    const int krow = k0 + lr;
    float kokf = 1.0f;
    int   kcl  = krow;
    if (GUARD) {
        kokf = (krow >= 0 && krow < SS) ? 1.0f : 0.0f;
        kcl  = krow < 0 ? 0 : (krow >= SS - 1 ? SS - 1 : krow);
    }
    const float* kp = kbase + (size_t)kcl * row + khalf;

    v8f acc = v8f{0,0,0,0,0,0,0,0};
#pragma unroll
    for (int c = 0; c < HDIM; c += 4) {
        v2f a = *(const v2f*)(kp + c);
        if (GUARD) a *= kokf;                    // branch-free OOB zeroing
        v2f bq = *(const v2f*)(qp + c);
        acc = wmma_f32(a, bq, acc);
    }
    return acc;
}

// ---------------------------------------------------------------------------
// One 16x16 out^T tile: V^T(16 x 144) @ P^T(144 x 16).
// ---------------------------------------------------------------------------
template <bool GUARD>
__device__ inline v8f out_tile(const float* __restrict__ vbase,
                               const float* __restrict__ PT,
                               int q0, int d0, int lr, int khalf, int row) {
    v8f acc = v8f{0,0,0,0,0,0,0,0};
    for (int c = 0; c < KLEN; c += 4) {
        const int cc  = c + khalf;
        const int ky0 = q0 - WIN + cc;
        const int ky1 = ky0 + 1;
        float a0, a1;
        if (GUARD) {
            const float f0 = (ky0 >= 0 && ky0 < SS) ? 1.0f : 0.0f;
            const float f1 = (ky1 >= 0 && ky1 < SS) ? 1.0f : 0.0f;
            const int  c0 = ky0 < 0 ? 0 : (ky0 >= SS - 1 ? SS - 1 : ky0);
            const int  c1 = ky1 < 0 ? 0 : (ky1 >= SS - 1 ? SS - 1 : ky1);
            a0 = vbase[(size_t)c0 * row + d0 + lr] * f0;
            a1 = vbase[(size_t)c1 * row + d0 + lr] * f1;
        } else {
            a0 = vbase[(size_t)ky0 * row + d0 + lr];
            a1 = vbase[(size_t)ky1 * row + d0 + lr];
        }
        v2f a  = v2f{a0, a1};
        v2f bp = v2f{PT[cc * 16 + lr], PT[(cc + 1) * 16 + lr]};
        acc = wmma_f32(a, bp, acc);
    }
    return acc;
}

// ---------------------------------------------------------------------------
// Sliding-window attention, one wave per (batch, head, 16-query tile).
// Scores computed transposed (S^T = K @ Q^T); softmax over keys is per-lane
// + one __shfl_xor(16). Probabilities staged in LDS; out^T = V^T @ P^T.
// Interior tiles (94%) take a guard-free fast path (wave-uniform branch).
// ---------------------------------------------------------------------------
__global__ __launch_bounds__(32)
void attn_window_kernel(const float* __restrict__ qkv,
                        const float* __restrict__ pmask,
                        float* __restrict__ attn_out) {
    __shared__ float PT[KLEN * 16];   // [key (144)][query (16)]

    const int lane = threadIdx.x;
    const int qt = blockIdx.x;
    const int h  = blockIdx.y;
    const int b  = blockIdx.z;
    const int q0 = qt * 16;

    const int row = 3 * DD;           // qkv token stride (floats)
    const float* qbase = qkv + (size_t)b * SS * row + 0 * DD + h * HDIM;
    const float* kbase = qkv + (size_t)b * SS * row + 1 * DD + h * HDIM;
    const float* vbase = qkv + (size_t)b * SS * row + 2 * DD + h * HDIM;

    const int lr    = lane & 15;
    const int khalf = (lane >> 4) << 1;    // 0 or 2
    const int hi8   = (lane >> 4) << 3;    // 0 or 8

    // wave-uniform: every candidate key of this q-tile is inside [0, S)
    const bool interior = (q0 - WIN >= 0) && (q0 + 15 + WIN < SS);

    // ---- scores: S^T[key][query], 9 key tiles ----
    const float* qp = qbase + (size_t)(q0 + lr) * row + khalf;
    for (int kt = 0; kt < KT; ++kt) {
        const int k0 = q0 - WIN + kt * 16;
        v8f acc = interior ? score_tile<false>(kbase, qp, k0, lr, khalf, row)
                           : score_tile<true >(kbase, qp, k0, lr, khalf, row);
#pragma unroll
        for (int v = 0; v < 8; ++v) {
            const int key = k0 + v + hi8;
            const int qq  = q0 + lr;
            int dk = key - qq; if (dk < 0) dk = -dk;
            const bool keyok = (interior || (key >= 0 && key < SS)) && dk <= WIN;
            const int  kcl   = key < 0 ? 0 : (key >= SS - 1 ? SS - 1 : key);
            const float val = keyok
                ? acc[v] * 0.125f + (1.0f - pmask[b * SS + kcl]) * NEGV
                : NEGV;                          // exp underflows to exactly 0
            PT[(kt * 16 + v + hi8) * 16 + lr] = val;
        }
    }
    __syncthreads();

    // ---- softmax over the 144 keys of each query column ----
    {
        const int col  = lane & 15;
        const int rbeg = (lane >> 4) * (KLEN / 2);
        const int rend = rbeg + (KLEN / 2);
        float m = -3.4e38f;
        for (int r = rbeg; r < rend; ++r) m = fmaxf(m, PT[r * 16 + col]);
        m = fmaxf(m, __shfl_xor(m, 16, 32));
        float s = 0.f;
        for (int r = rbeg; r < rend; ++r) {
            float e = __expf(PT[r * 16 + col] - m);
            PT[r * 16 + col] = e;
            s += e;
        }
        s += __shfl_xor(s, 16, 32);
        const float inv = 1.0f / s;
        for (int r = rbeg; r < rend; ++r) PT[r * 16 + col] *= inv;
    }
    __syncthreads();

    // ---- out^T = V^T(64 x 144) @ P^T(144 x 16), 4 head-dim tiles ----
#pragma unroll
    for (int dt = 0; dt < 4; ++dt) {
        const int d0 = dt * 16;
        v8f acc = interior ? out_tile<false>(vbase, PT, q0, d0, lr, khalf, row)
                           : out_tile<true >(vbase, PT, q0, d0, lr, khalf, row);
#pragma unroll
        for (int v = 0; v < 8; ++v) {
            const int d  = d0 + v + hi8;
            const int qq = q0 + lr;
            attn_out[((size_t)b * SS + qq) * DD + h * HDIM + d] = acc[v];
        }
    }
}

// ---------------------------------------------------------------------------
extern "C" void kernel_launch(void* const* d_in, const int* in_sizes, int n_in,
                              void* d_out, int out_size, void* d_ws, size_t ws_size,
                              hipStream_t stream) {
    const float* x    = (const float*)d_in[0];   // (B,S,D)
    const float* pm   = (const float*)d_in[1];   // (B,S)
    const float* Wqkv = (const float*)d_in[2];   // (D, 3D)
    const float* Wo   = (const float*)d_in[3];   // (D, D)
    float* out = (float*)d_out;                  // (B,S,D)

    float* qkv  = (float*)d_ws;                          // B*S*3*D floats
    float* attn = qkv + (size_t)BB * SS * 3 * DD;        // B*S*D floats

    const int M = BB * SS;                       // 4096 tokens

    // 1) qkv = x @ Wqkv   (4096 x 768) @ (768 x 2304)
    {
        dim3 grid((3 * DD) / 64, M / 128);
        wmma_gemm_f32<2, 4><<<grid, dim3(128), 0, stream>>>(x, Wqkv, qkv, M, 3 * DD, DD);
    }
    // 2) windowed attention
    {
        dim3 grid(SS / 16, HH, BB);
        attn_window_kernel<<<grid, dim3(32), 0, stream>>>(qkv, pm, attn);
    }
    // 3) out = attn @ Wo   (4096 x 768) @ (768 x 768)
    {
        dim3 grid(DD / 64, M / 128);
        wmma_gemm_f32<2, 4><<<grid, dim3(128), 0, stream>>>(attn, Wo, out, M, DD, DD);
    }
}